// CostomAdaptiveAvgPool2D_24412594111127
// MI455X (gfx1250) — compile-verified
//
#include <hip/hip_runtime.h>

// ---------------------------------------------------------------------------
// Adaptive avg-pool (16,768,64,48) f32 -> (16,768,7,7) as two WMMA GEMMs:
//   Out = Mh(7x64,pad16) * X(64x48) * Mw^T(48x7,pad16)
// Memory-bound: 151 MB / 23.3 TB/s ~= 6.5 us floor. To stay under that floor
// on the instruction-issue side, each wave processes 8 channels and the
// constant pooling-matrix operands are synthesized ONCE per wave (hoisted
// into registers), leaving a per-channel body of ~96 coalesced loads,
// 60 v_wmma_f32_16x16x4_f32, and a tiny LDS transpose.
// ---------------------------------------------------------------------------

typedef __attribute__((ext_vector_type(2))) float v2f;
typedef __attribute__((ext_vector_type(8))) float v8f;

#define H_IN   64
#define W_IN   48
#define H_OUT  7
#define W_OUT  7
#define CH_TOTAL (16 * 768)
#define WAVES_PER_BLOCK 8
#define CH_PER_WAVE 8
#define BLOCKS (CH_TOTAL / (WAVES_PER_BLOCK * CH_PER_WAVE))   // 192
#define LDS_STRIDE 49   // coprime with 64 banks -> conflict-free A-layout reads

__global__ void __launch_bounds__(256)
adaptive_pool_wmma_kernel(const float* __restrict__ x,
                          float* __restrict__ out) {
    __shared__ float lds[WAVES_PER_BLOCK][16 * LDS_STRIDE];

    const int lane = threadIdx.x & 31;
    // Force wave index scalar so channel/LDS bases live in SGPRs.
    const int wv   = __builtin_amdgcn_readfirstlane((int)(threadIdx.x >> 5));

    const int row   = lane & 15;        // A-layout M index
    const int col   = lane & 15;        // B/C-layout N index
    const int khalf = (lane >> 4) * 2;  // upper half-wave holds K+2,K+3

    // ---- hoisted stage-1 A operand: Mh_pad[row][*] for all 16 K-steps -----
    const int hs = (row * H_IN) / H_OUT;
    const int he = ((row + 1) * H_IN + H_OUT - 1) / H_OUT;
    const float hinv = (row < H_OUT) ? 1.0f / (float)(he - hs) : 0.0f;
    v2f a_arr[16];
#pragma unroll
    for (int ks = 0; ks < 16; ++ks) {
        const int ka = ks * 4 + khalf;
        a_arr[ks].x = (ka     >= hs && ka     < he) ? hinv : 0.0f;
        a_arr[ks].y = (ka + 1 >= hs && ka + 1 < he) ? hinv : 0.0f;
    }

    // ---- hoisted stage-2 B operand: Mw^T_pad[*][col] for all 12 K-steps ---
    const int wws = (col * W_IN) / W_OUT;
    const int wwe = ((col + 1) * W_IN + W_OUT - 1) / W_OUT;
    const float winv = (col < W_OUT) ? 1.0f / (float)(wwe - wws) : 0.0f;
    v2f b2_arr[12];
#pragma unroll
    for (int ks = 0; ks < 12; ++ks) {
        const int kb = ks * 4 + khalf;
        b2_arr[ks].x = (kb     >= wws && kb     < wwe) ? winv : 0.0f;
        b2_arr[ks].y = (kb + 1 >= wws && kb + 1 < wwe) ? winv : 0.0f;
    }

    const int ch0 = (blockIdx.x * WAVES_PER_BLOCK + wv) * CH_PER_WAVE;
    const int in_lane_off = khalf * W_IN + col;   // per-lane constant offset
    float* tl = &lds[wv][0];

    for (int i = 0; i < CH_PER_WAVE; ++i) {
        const int ch = ch0 + i;
        const float* __restrict__ Xc = x + (size_t)ch * (H_IN * W_IN);
        const float* __restrict__ Xl = Xc + in_lane_off;

        // Light prefetch of the next channel (global_prefetch_b8).
        if (i + 1 < CH_PER_WAVE)
            __builtin_prefetch(Xc + H_IN * W_IN + lane * 96, 0, 0);

        // ---- Stage 1: T(16x48) = Mh_pad * X; 96 coalesced b32 loads -------
        v8f acc0 = {}, acc1 = {}, acc2 = {};
#pragma unroll
        for (int ks = 0; ks < 16; ++ks) {
            const int base = ks * 4 * W_IN;   // compile-time immediates
            v2f b0, b1, b2;
            b0.x = Xl[base     ];  b0.y = Xl[base + W_IN     ];
            b1.x = Xl[base + 16];  b1.y = Xl[base + W_IN + 16];
            b2.x = Xl[base + 32];  b2.y = Xl[base + W_IN + 32];

            acc0 = __builtin_amdgcn_wmma_f32_16x16x4_f32(false, a_arr[ks], false, b0,
                                                         (short)0, acc0, false, false);
            acc1 = __builtin_amdgcn_wmma_f32_16x16x4_f32(false, a_arr[ks], false, b1,
                                                         (short)0, acc1, false, false);
            acc2 = __builtin_amdgcn_wmma_f32_16x16x4_f32(false, a_arr[ks], false, b2,
                                                         (short)0, acc2, false, false);
        }

        // ---- Spill T (C-layout) to LDS row-major [m][w] -------------------
        {
            const int m_lo = (lane >> 4) * 8;  // lanes 0-15: m=r ; 16-31: m=r+8
#pragma unroll
            for (int r = 0; r < 8; ++r) {
                const int m = m_lo + r;
                tl[m * LDS_STRIDE + col +  0] = acc0[r];
                tl[m * LDS_STRIDE + col + 16] = acc1[r];
                tl[m * LDS_STRIDE + col + 32] = acc2[r];
            }
        }
        __syncthreads();   // uniform trip count; fences LDS store -> load

        // ---- Stage 2: Out = T * Mw^T_pad ----------------------------------
        v8f oacc = {};
#pragma unroll
        for (int ks = 0; ks < 12; ++ks) {
            const int ka = ks * 4 + khalf;
            v2f a;
            a.x = tl[row * LDS_STRIDE + ka];
            a.y = tl[row * LDS_STRIDE + ka + 1];
            oacc = __builtin_amdgcn_wmma_f32_16x16x4_f32(false, a, false, b2_arr[ks],
                                                         (short)0, oacc, false, false);
        }

        // ---- Store valid 7x7: lanes 0-6 (lower half), VGPRs 0-6 -----------
        float* __restrict__ O = out + (size_t)ch * (H_OUT * W_OUT);
        if (lane < W_OUT) {
#pragma unroll
            for (int r = 0; r < H_OUT; ++r)
                O[r * W_OUT + lane] = oacc[r];
        }
        __syncthreads();   // WAR: finish reading LDS before next iteration
    }
}

extern "C" void kernel_launch(void* const* d_in, const int* in_sizes, int n_in,
                              void* d_out, int out_size, void* d_ws, size_t ws_size,
                              hipStream_t stream) {
    (void)in_sizes; (void)n_in; (void)out_size; (void)d_ws; (void)ws_size;
    const float* x = (const float*)d_in[0];
    float* out     = (float*)d_out;

    adaptive_pool_wmma_kernel<<<BLOCKS, 256, 0, stream>>>(x, out);
}